// GCN_10282151707565
// MI455X (gfx1250) — compile-verified
//
#include <hip/hip_runtime.h>

typedef float v2f __attribute__((ext_vector_type(2)));
typedef float v8f __attribute__((ext_vector_type(8)));

// ---------------------------------------------------------------------------
// Degree init: deg[i] = 1.0f (self-loop contribution)
// ---------------------------------------------------------------------------
__global__ void gcn_deg_init(float* __restrict__ deg, int n) {
    int i = blockIdx.x * blockDim.x + threadIdx.x;
    if (i < n) deg[i] = 1.0f;
}

// deg[dst[e]] += 1 for every edge
__global__ void gcn_deg_accum(const long long* __restrict__ dst,
                              float* __restrict__ deg, int e) {
    int i = blockIdx.x * blockDim.x + threadIdx.x;
    if (i < e) atomicAdd(&deg[(int)dst[i]], 1.0f);
}

// in-place: deg -> rsqrt(deg)   (deg >= 1 always due to self-loop)
__global__ void gcn_dinv(float* __restrict__ deg, int n) {
    int i = blockIdx.x * blockDim.x + threadIdx.x;
    if (i < n) deg[i] = rsqrtf(deg[i]);
}

// ---------------------------------------------------------------------------
// Dense GEMM  C[nRows x NCOLS] = A[nRows x 128] * B[128 x NCOLS]
// One wave32 per 16x16 output tile, fully unrolled K-loop of
// V_WMMA_F32_16X16X4_F32 (32 WMMAs). NCOLS/K are compile-time so all loads
// become base + immediate-offset; no exec predication in the loop (B column
// is clamped instead of guarded: column n of C depends only on column n of B,
// and out-of-range columns are never stored).
//
// ISA f32 layouts (CDNA5 ISA):
//   A 16x4 : lanes 0-15 -> M, VGPR0 = K{0|2}, VGPR1 = K{1|3}  (kh = lane>>4)
//   B 4x16 : lanes 0-15 -> N, VGPR0 = K{0|2}, VGPR1 = K{1|3}
//   C 16x16: VGPR i -> M = i + 8*(lane>>4), N = lane&15
// ---------------------------------------------------------------------------
template <int NCOLS>
__global__ void gcn_gemm_wmma_f32(const float* __restrict__ A,
                                  const float* __restrict__ B,
                                  float* __restrict__ C,
                                  int nRows) {
    constexpr int K = 128;
    constexpr int colTiles = (NCOLS + 15) >> 4;
    const int totalTiles = (nRows >> 4) * colTiles;

    // tile id is wave-uniform: pin it to an SGPR so row/col/address bases are
    // scalar and EXEC stays all-ones for the WMMAs.
    int wid = (int)(((long long)blockIdx.x * blockDim.x + threadIdx.x) >> 5);
    wid = __builtin_amdgcn_readfirstlane(wid);
    if (wid >= totalTiles) return;

    const int lane = threadIdx.x & 31;
    const int m    = lane & 15;     // A row in tile / B-C col in tile
    const int kh   = lane >> 4;     // which K-half this lane holds

    const int row = (wid / colTiles) << 4;
    const int col = (wid % colTiles) << 4;

    const int bcol  = col + m;                               // desired C col
    const int bcolc = bcol < NCOLS ? bcol : (NCOLS - 1);     // clamped (safe)

    const float2* a2 = reinterpret_cast<const float2*>(
        A + (size_t)(row + m) * K + kh * 2);                 // 8B aligned
    const float* bp = B + bcolc;

    v8f acc = {};

    #pragma unroll
    for (int k = 0; k < K; k += 4) {
        float2 av = a2[k >> 1];        // A[(row+m)*K + k + kh*2 + {0,1}]
        v2f a; a.x = av.x; a.y = av.y;

        const int kb = k + kh * 2;
        v2f b;
        b.x = bp[(size_t)kb * NCOLS];
        b.y = bp[(size_t)(kb + 1) * NCOLS];

        acc = __builtin_amdgcn_wmma_f32_16x16x4_f32(
            /*neg_a=*/false, a, /*neg_b=*/false, b,
            /*c_mod=*/(short)0, acc, /*reuse_a=*/false, /*reuse_b=*/false);
    }

    if ((NCOLS & 15) == 0 || bcol < NCOLS) {
        #pragma unroll
        for (int i = 0; i < 8; ++i) {
            int r = row + kh * 8 + i;
            C[(size_t)r * NCOLS + bcol] = acc[i];
        }
    }
}

// ---------------------------------------------------------------------------
// Edge scatter, F=128: one wave per edge, 32 lanes x float4 coalesced gather,
// 4 x global_atomic_add_f32 per lane into agg[dst]. src/dst/w are wave-
// uniform -> readfirstlane keeps the address math scalar.
// ---------------------------------------------------------------------------
__global__ void gcn_scatter128(const float* __restrict__ h,
                               const float* __restrict__ dinv,
                               const long long* __restrict__ src,
                               const long long* __restrict__ dst,
                               float* __restrict__ agg, int e) {
    int wid = (int)(((long long)blockIdx.x * blockDim.x + threadIdx.x) >> 5);
    wid = __builtin_amdgcn_readfirstlane(wid);
    int lane = threadIdx.x & 31;
    if (wid >= e) return;
    int s = __builtin_amdgcn_readfirstlane((int)src[wid]);
    int d = __builtin_amdgcn_readfirstlane((int)dst[wid]);
    float w = dinv[s] * dinv[d];
    const float4* hp = reinterpret_cast<const float4*>(h + (size_t)s * 128);
    float4 v = hp[lane];
    float* ap = agg + (size_t)d * 128 + lane * 4;
    atomicAdd(ap + 0, v.x * w);
    atomicAdd(ap + 1, v.y * w);
    atomicAdd(ap + 2, v.z * w);
    atomicAdd(ap + 3, v.w * w);
}

// Edge scatter, generic F (used with F=40 for layer 2)
__global__ void gcn_scatterF(const float* __restrict__ g,
                             const float* __restrict__ dinv,
                             const long long* __restrict__ src,
                             const long long* __restrict__ dst,
                             float* __restrict__ agg, int e, int F) {
    int wid = (int)(((long long)blockIdx.x * blockDim.x + threadIdx.x) >> 5);
    wid = __builtin_amdgcn_readfirstlane(wid);
    int lane = threadIdx.x & 31;
    if (wid >= e) return;
    int s = __builtin_amdgcn_readfirstlane((int)src[wid]);
    int d = __builtin_amdgcn_readfirstlane((int)dst[wid]);
    float w = dinv[s] * dinv[d];
    const float* gp = g + (size_t)s * F;
    float* ap = agg + (size_t)d * F;
    for (int f = lane; f < F; f += 32)
        atomicAdd(ap + f, gp[f] * w);
}

// layer-1 epilogue: agg <- relu(agg + dinv^2 * h + b1)   (in-place into agg)
__global__ void gcn_finalize_relu(float* __restrict__ agg,
                                  const float* __restrict__ h,
                                  const float* __restrict__ dinv,
                                  const float* __restrict__ bias,
                                  int n, int F) {
    long long i = (long long)blockIdx.x * blockDim.x + threadIdx.x;
    if (i >= (long long)n * F) return;
    int node = (int)(i / F);
    int f    = (int)(i % F);
    float di = dinv[node];
    float v = agg[i] + di * di * h[i] + bias[f];
    agg[i] = v > 0.0f ? v : 0.0f;
}

// layer-2 epilogue: out = agg + dinv^2 * g + b2
__global__ void gcn_finalize(const float* __restrict__ agg,
                             const float* __restrict__ g,
                             const float* __restrict__ dinv,
                             const float* __restrict__ bias,
                             float* __restrict__ out, int n, int F) {
    long long i = (long long)blockIdx.x * blockDim.x + threadIdx.x;
    if (i >= (long long)n * F) return;
    int node = (int)(i / F);
    int f    = (int)(i % F);
    float di = dinv[node];
    out[i] = agg[i] + di * di * g[i] + bias[f];
}

// ---------------------------------------------------------------------------
extern "C" void kernel_launch(void* const* d_in, const int* in_sizes, int n_in,
                              void* d_out, int out_size, void* d_ws, size_t ws_size,
                              hipStream_t stream) {
    const float*     x  = (const float*)d_in[0];
    const long long* ei = (const long long*)d_in[1];   // int64 per reference
    const float*     W1 = (const float*)d_in[2];
    const float*     b1 = (const float*)d_in[3];
    const float*     W2 = (const float*)d_in[4];
    const float*     b2 = (const float*)d_in[5];

    const int F1 = 128;
    const int N  = in_sizes[0] / F1;           // 100000
    const int E  = in_sizes[1] / 2;            // 1600000
    const int F2 = in_sizes[5];                // 40 (num_classes)

    const long long* src = ei;
    const long long* dst = ei + E;

    // workspace layout (bytes, 256-aligned)
    char* ws = (char*)d_ws;
    auto align256 = [](size_t v) { return (v + 255) & ~(size_t)255; };
    size_t degOff  = 0;
    size_t hOff    = align256(degOff + (size_t)N * 4);
    size_t agg1Off = align256(hOff + (size_t)N * F1 * 4);
    // after layer-1 epilogue, h region is dead -> reuse it for g and agg2
    size_t gOff    = hOff;
    size_t agg2Off = align256(gOff + (size_t)N * F2 * 4);

    float* deg  = (float*)(ws + degOff);   // becomes dinv in-place
    float* h    = (float*)(ws + hOff);
    float* agg1 = (float*)(ws + agg1Off); // becomes h2 (relu) in-place
    float* g    = (float*)(ws + gOff);
    float* agg2 = (float*)(ws + agg2Off);
    float* out  = (float*)d_out;

    const int TB = 256;
    auto blocks = [&](long long work) { return (int)((work + TB - 1) / TB); };

    // ---- normalization ----
    gcn_deg_init<<<blocks(N), TB, 0, stream>>>(deg, N);
    hipMemsetAsync(agg1, 0, (size_t)N * F1 * 4, stream);
    gcn_deg_accum<<<blocks(E), TB, 0, stream>>>(dst, deg, E);
    gcn_dinv<<<blocks(N), TB, 0, stream>>>(deg, N);

    // ---- layer 1: h = x @ W1 ----
    {
        long long tiles = (long long)(N / 16) * (F1 / 16);
        gcn_gemm_wmma_f32<128><<<blocks(tiles * 32), TB, 0, stream>>>(x, W1, h, N);
    }
    gcn_scatter128<<<blocks((long long)E * 32), TB, 0, stream>>>(h, deg, src, dst, agg1, E);
    gcn_finalize_relu<<<blocks((long long)N * F1), TB, 0, stream>>>(agg1, h, deg, b1, N, F1);

    // ---- layer 2: g = h2 @ W2 ----
    hipMemsetAsync(agg2, 0, (size_t)N * F2 * 4, stream);
    {
        long long tiles = (long long)(N / 16) * ((F2 + 15) / 16);
        gcn_gemm_wmma_f32<40><<<blocks(tiles * 32), TB, 0, stream>>>(agg1, W2, g, N);
    }
    gcn_scatterF<<<blocks((long long)E * 32), TB, 0, stream>>>(g, deg, src, dst, agg2, E, F2);
    gcn_finalize<<<blocks((long long)N * F2), TB, 0, stream>>>(agg2, g, deg, b2, out, N, F2);
}